// VideoEncoder_25838523253483
// MI455X (gfx1250) — compile-verified
//
#include <hip/hip_runtime.h>
#include <hip/hip_bf16.h>

typedef __attribute__((ext_vector_type(2))) float v2f;
typedef __attribute__((ext_vector_type(8))) float v8f;

// ---------------------------------------------------------------------------
// Kernel 1: conv1 (Cin=3, K=81) direct + ReLU.  video[2,3,16,128,128] (NCDHW)
// -> h1 [2,16,64,64,128] (NDHWC, channels-last).  co is the fastest index so
// stores are coalesced and spatial input reads are wave-uniform broadcasts.
// ---------------------------------------------------------------------------
__global__ void conv1_relu_kernel(const float* __restrict__ x,
                                  const float* __restrict__ w,
                                  const float* __restrict__ bias,
                                  float* __restrict__ y) {
  int idx = blockIdx.x * 256 + threadIdx.x;       // 2*16*64*64*128 total
  int co = idx & 127;
  int t1 = idx >> 7;
  int wo = t1 & 63; t1 >>= 6;
  int ho = t1 & 63; t1 >>= 6;
  int t  = t1 & 15;
  int n  = t1 >> 4;
  float acc = bias[co];
  #pragma unroll
  for (int ci = 0; ci < 3; ++ci) {
    for (int kd = 0; kd < 3; ++kd) {
      int ti = t + kd - 1;
      if ((unsigned)ti >= 16u) continue;
      for (int kh = 0; kh < 3; ++kh) {
        int hi = 2 * ho + kh - 1;
        if ((unsigned)hi >= 128u) continue;
        for (int kw = 0; kw < 3; ++kw) {
          int wi = 2 * wo + kw - 1;
          if ((unsigned)wi >= 128u) continue;
          acc += x[((size_t)(n * 3 + ci) * 16 + ti) * 16384 + hi * 128 + wi] *
                 w[((co * 3 + ci) * 3 + kd) * 9 + kh * 3 + kw];
        }
      }
    }
  }
  y[idx] = fmaxf(acc, 0.0f);   // NDHWC layout == idx decomposition above
}

// ---------------------------------------------------------------------------
// Kernel 2: weight repack  [co][ci][tap] (OIDHW) -> [tap][ci][co].
// Makes conv WMMA B-tile loads coalesced across lanes (16 consecutive co).
// ---------------------------------------------------------------------------
__global__ void repack_w_kernel(const float* __restrict__ w,
                                float* __restrict__ wp, int Cout, int Cin) {
  int idx = blockIdx.x * 256 + threadIdx.x;       // Cout*Cin*27 total
  int co  = idx % Cout;
  int tmp = idx / Cout;
  int ci  = tmp % Cin;
  int tap = tmp / Cin;
  wp[idx] = w[(size_t)(co * Cin + ci) * 27 + tap];
}

// ---------------------------------------------------------------------------
// Kernels 3/4: conv2/conv3 as tap-decomposed implicit GEMM on
// V_WMMA_F32_16X16X4_F32, channels-last in and out.  One wave per block.
//   M tile = 16 consecutive wo at fixed (n, t, ho); N tile = 16 out channels.
//   K = 27 taps x (CIN/4) WMMA steps.
// A layout: M=lane%16, K=(lane>=16?2:0)+reg -> adjacent c pair = b64 load.
// B layout: N=lane%16 -> 16 consecutive co per load (coalesced).
// D layout: M = reg + (lane>=16)*8, N = lane%16.
// ---------------------------------------------------------------------------
template <int CIN, int HIN, int HOUT, bool RELU>
__global__ void conv_wmma_kernel(const float* __restrict__ x,
                                 const float* __restrict__ wp,
                                 const float* __restrict__ bias,
                                 float* __restrict__ y, int Cout) {
  const int lane = threadIdx.x;        // blockDim.x == 32, one wave
  const int m    = lane & 15;
  const int half = lane >> 4;
  const int k0   = half * 2;

  int mt = blockIdx.x;
  const int WT = HOUT / 16;
  int wt = mt % WT; mt /= WT;
  int ho = mt % HOUT;
  int t  = mt / HOUT;
  const int co0 = blockIdx.y * 16;
  const int n   = blockIdx.z;
  const int wo  = wt * 16 + m;

  v8f acc = {};
  for (int kd = 0; kd < 3; ++kd) {
    int ti = t + kd - 1;
    if ((unsigned)ti >= 16u) continue;              // wave-uniform
    for (int kh = 0; kh < 3; ++kh) {
      int hi = 2 * ho + kh - 1;
      if ((unsigned)hi >= (unsigned)HIN) continue;  // wave-uniform
      const float* xrow =
          x + (((size_t)(n * 16 + ti) * HIN + hi) * HIN) * CIN;
      for (int kw = 0; kw < 3; ++kw) {
        int wi = 2 * wo + kw - 1;                   // per-lane
        bool inb = (unsigned)wi < (unsigned)HIN;
        const float* xp = xrow + (size_t)wi * CIN;  // contiguous c
        int tap = (kd * 3 + kh) * 3 + kw;
        const float* bp = wp + (size_t)tap * CIN * Cout + co0 + m;
        for (int c0 = 0; c0 < CIN; c0 += 4) {
          int cA = c0 + k0;
          v2f a, b;
          a.x = inb ? xp[cA] : 0.0f;                // adjacent pair -> b64
          a.y = inb ? xp[cA + 1] : 0.0f;
          b.x = bp[(size_t)cA * Cout];              // coalesced across lanes
          b.y = bp[(size_t)(cA + 1) * Cout];
          acc = __builtin_amdgcn_wmma_f32_16x16x4_f32(
              false, a, false, b, (short)0, acc, false, false);
        }
      }
    }
  }

  const int co = co0 + m;                 // N index for this lane
  const float bv = bias[co];
  // channels-last store: ((((n*16+t)*HOUT+ho)*HOUT + wo)*Cout) + co
  size_t base = (((size_t)(n * 16 + t) * HOUT + ho) * HOUT + (size_t)wt * 16) *
                    Cout + co;
  #pragma unroll
  for (int r = 0; r < 8; ++r) {
    int mm = r + half * 8;                // M (spatial) index
    float v = acc[r] + bv;
    if (RELU) v = fmaxf(v, 0.0f);
    y[base + (size_t)mm * Cout] = v;
  }
}

// ---------------------------------------------------------------------------
// Kernel 5: codebook squared norms (8 rows per block, wave per row).
// ---------------------------------------------------------------------------
__global__ void codebook_norms_kernel(const float* __restrict__ cb,
                                      float* __restrict__ c2) {
  int row  = blockIdx.x * 8 + (threadIdx.x >> 5);
  int lane = threadIdx.x & 31;
  const float* p = cb + (size_t)row * 512;
  float s = 0.0f;
  for (int c = lane; c < 512; c += 32) { float v = p[c]; s += v * v; }
  #pragma unroll
  for (int off = 16; off > 0; off >>= 1) s += __shfl_xor(s, off, 32);
  if (lane == 0) c2[row] = s;
}

// ---------------------------------------------------------------------------
// Kernel 6: VQ argmin.  feats are channels-last: f[(b*4096+pos)*512 + c].
// Block = 256 threads (8 waves), pos-tile of 16.  Feats tile staged in LDS
// via GLOBAL_LOAD_ASYNC_TO_LDS_B128 (ASYNCcnt path, no VGPR bounce; row
// stride 516 floats = 2064 B keeps 16-B alignment and spreads banks); each
// wave covers a strided subset of the 512 codebook v-tiles with a K=512 f32
// WMMA GEMM; per-lane running best -> shfl_xor half-wave reduce -> LDS
// cross-wave reduce -> token.
// argmin(||f||^2 + ||c||^2 - 2 f.c) == argmin(c2[v] - 2*dot).
// ---------------------------------------------------------------------------
#define VQ_WAVES 8
__global__ void vq_argmin_kernel(const float* __restrict__ f,
                                 const float* __restrict__ cb,
                                 const float* __restrict__ c2,
                                 float* __restrict__ tok_f,
                                 int* __restrict__ tok_i) {
  __shared__ float As[16 * 516];
  __shared__ float redS[VQ_WAVES * 16];
  __shared__ int   redI[VQ_WAVES * 16];

  const int tid  = threadIdx.x;
  const int lane = tid & 31;
  const int wv   = tid >> 5;
  const int b    = blockIdx.y;
  const int pos0 = blockIdx.x * 16;
  const float* fb = f + ((size_t)b * 4096 + pos0) * 512;

  // Async global->LDS staging: 16 rows x 512 floats = 2048 x b128 chunks.
  for (int ch = tid; ch < 16 * 128; ch += 256) {
    int row = ch >> 7;                 // 128 b128-chunks per row
    int c4  = (ch & 127) << 2;         // float index, 16B aligned
    unsigned ldsOff = (unsigned)(uintptr_t)(&As[row * 516 + c4]);
    unsigned long long gaddr =
        (unsigned long long)(uintptr_t)(fb + (size_t)row * 512 + c4);
    asm volatile("global_load_async_to_lds_b128 %0, %1, off"
                 :: "v"(ldsOff), "v"(gaddr) : "memory");
  }
  asm volatile("s_wait_asynccnt 0x0" ::: "memory");
  __syncthreads();

  const int m    = lane & 15;
  const int half = lane >> 4;
  const int k0   = half * 2;

  float bestS[8];
  int   bestI[8];
  #pragma unroll
  for (int r = 0; r < 8; ++r) { bestS[r] = 3.0e38f; bestI[r] = 0x7fffffff; }

  const float* ap = &As[m * 516];
  for (int vt = wv; vt < 512; vt += VQ_WAVES) {
    int v0 = vt * 16;
    const float* bp = cb + (size_t)(v0 + m) * 512;
    v8f acc = {};
    for (int c0 = 0; c0 < 512; c0 += 4) {
      v2f a, bv;
      a.x  = ap[c0 + k0];
      a.y  = ap[c0 + k0 + 1];
      bv.x = bp[c0 + k0];
      bv.y = bp[c0 + k0 + 1];
      acc = __builtin_amdgcn_wmma_f32_16x16x4_f32(
          false, a, false, bv, (short)0, acc, false, false);
    }
    float cn  = c2[v0 + m];     // N index of this lane's D column
    int vidx  = v0 + m;
    #pragma unroll
    for (int r = 0; r < 8; ++r) {
      float s = cn - 2.0f * acc[r];
      if (s < bestS[r] || (s == bestS[r] && vidx < bestI[r])) {
        bestS[r] = s; bestI[r] = vidx;
      }
    }
  }

  // Reduce over the 16 lanes of each half (they share the same pos per reg).
  #pragma unroll
  for (int off = 1; off <= 8; off <<= 1) {
    #pragma unroll
    for (int r = 0; r < 8; ++r) {
      float s2 = __shfl_xor(bestS[r], off, 32);
      int   i2 = __shfl_xor(bestI[r], off, 32);
      if (s2 < bestS[r] || (s2 == bestS[r] && i2 < bestI[r])) {
        bestS[r] = s2; bestI[r] = i2;
      }
    }
  }
  if ((lane & 15) == 0) {
    #pragma unroll
    for (int r = 0; r < 8; ++r) {
      int p = r + half * 8;
      redS[wv * 16 + p] = bestS[r];
      redI[wv * 16 + p] = bestI[r];
    }
  }
  __syncthreads();
  if (tid < 16) {
    float s = redS[tid];
    int  vi = redI[tid];
    for (int w2 = 1; w2 < VQ_WAVES; ++w2) {
      float s2 = redS[w2 * 16 + tid];
      int   i2 = redI[w2 * 16 + tid];
      if (s2 < s || (s2 == s && i2 < vi)) { s = s2; vi = i2; }
    }
    tok_f[(size_t)b * 4096 + pos0 + tid] = (float)vi;
    tok_i[b * 4096 + pos0 + tid] = vi;
  }
}

// ---------------------------------------------------------------------------
// Kernel 7: embedding gather (one block per (b,pos), coalesced 512-float row).
// ---------------------------------------------------------------------------
__global__ void gather_emb_kernel(const int* __restrict__ tok,
                                  const float* __restrict__ emb,
                                  float* __restrict__ out) {
  int pos = blockIdx.x;                 // 0..8191 (= b*4096 + pos)
  int t = tok[pos];
  const float* src = emb + (size_t)t * 512;
  float* dst = out + (size_t)pos * 512;
  for (int c = threadIdx.x; c < 512; c += blockDim.x) dst[c] = src[c];
}

// ---------------------------------------------------------------------------
extern "C" void kernel_launch(void* const* d_in, const int* in_sizes, int n_in,
                              void* d_out, int out_size, void* d_ws,
                              size_t ws_size, hipStream_t stream) {
  const float* video    = (const float*)d_in[0];  // [2,3,16,128,128]
  const float* w1       = (const float*)d_in[1];  // [128,3,3,3,3]
  const float* b1       = (const float*)d_in[2];
  const float* w2       = (const float*)d_in[3];  // [256,128,3,3,3]
  const float* b2       = (const float*)d_in[4];
  const float* w3       = (const float*)d_in[5];  // [512,256,3,3,3]
  const float* b3       = (const float*)d_in[6];
  const float* codebook = (const float*)d_in[7];  // [8192,512]
  const float* emb      = (const float*)d_in[8];  // [8192,512]
  float* out = (float*)d_out;

  // Workspace layout (floats).  All activations channels-last.
  float* ws   = (float*)d_ws;
  float* h1   = ws;                          // [2,16,64,64,128] = 16,777,216
  float* h2   = h1 + 16777216;               // [2,16,32,32,256] =  8,388,608
  float* fflt = h2 + 4194304;                // [2,4096,512]     =  4,194,304
  float* c2   = fflt + 4194304;              // 8192
  int*   toki = (int*)(c2 + 8192);           // 8192 ints
  float* w2p  = (float*)(toki + 8192);       // [27][128][256] =    884,736
  float* w3p  = w2p + 884736;                // [27][256][512] =  3,538,944

  // 1) conv1 + relu (direct, NDHWC out)
  conv1_relu_kernel<<<65536, 256, 0, stream>>>(video, w1, b1, h1);

  // 2) weight repacks -> [tap][ci][co]
  repack_w_kernel<<<884736 / 256, 256, 0, stream>>>(w2, w2p, 256, 128);
  repack_w_kernel<<<3538944 / 256, 256, 0, stream>>>(w3, w3p, 512, 256);

  // 3) conv2 + relu (WMMA implicit GEMM): Hout=32 -> grid.x = 16*32*2 = 1024
  {
    dim3 grid(1024, 256 / 16, 2), block(32);
    conv_wmma_kernel<128, 64, 32, true>
        <<<grid, block, 0, stream>>>(h1, w2p, b2, h2, 256);
  }
  // 4) conv3 (WMMA implicit GEMM): Hout=16 -> grid.x = 16*16*1 = 256
  {
    dim3 grid(256, 512 / 16, 2), block(32);
    conv_wmma_kernel<256, 32, 16, false>
        <<<grid, block, 0, stream>>>(h2, w3p, b3, fflt, 512);
  }
  // 5) codebook norms
  codebook_norms_kernel<<<1024, 256, 0, stream>>>(codebook, c2);
  // 6) VQ argmin (WMMA GEMM + reduction): tokens -> d_out[0..8191] + ws ints
  {
    dim3 grid(256, 2), block(256);
    vq_argmin_kernel<<<grid, block, 0, stream>>>(fflt, codebook, c2, out, toki);
  }
  // 7) embedding gather -> d_out + 8192
  gather_emb_kernel<<<8192, 256, 0, stream>>>(toki, emb, out + 8192);
}